// PPMIGNN_71098888618091
// MI455X (gfx1250) — compile-verified
//
#include <hip/hip_runtime.h>

typedef __attribute__((ext_vector_type(2))) float v2f;
typedef __attribute__((ext_vector_type(4))) float v4f;
typedef __attribute__((ext_vector_type(8))) float v8f;

#define N_NODES 50000
#define FEAT 256
#define HID 128
#define EMB 64

#define AP 4   // LDS pad (floats) for A rows: stride % 64 banks == 4 -> conflict-free b64 frags
#define WP 8   // LDS pad (floats) for W rows: kh half-waves land on disjoint bank sets

// ---- CDNA5 async global->LDS copy (ASYNCcnt-tracked), per CDNA5 ISA ----
__device__ __forceinline__ void async_copy_b128(unsigned lds_off, const float* gsrc) {
    asm volatile("global_load_async_to_lds_b128 %0, %1, off"
                 :: "v"(lds_off), "v"(gsrc)
                 : "memory");
}
__device__ __forceinline__ void wait_asynccnt0() {
    asm volatile("s_wait_asynccnt 0x0" ::: "memory");
}

// ---------------------------------------------------------------------------
// Dense GEMM  H[M x N] = A[M x K] @ W[K x N]  (row-major fp32, WMMA f32 K=4)
// One block per 64-row M strip. Whole W staged in padded LDS once per block;
// A strip staged via async global->LDS b128. N/16 waves; each wave owns one
// 16-col N tile and MT=4 M tiles (4 independent accumulators reuse each B
// fragment, 4x less LDS-B traffic, deep XDL pipeline).
// f32 16x16x4 fragment layout (ISA 7.12.2):
//   A: lane={m=lane&15, kh=lane>>4}; a = A[m][k+2kh .. +1]
//   B: lane={n=lane&15, kh=lane>>4}; b = { B[k+2kh][n], B[k+2kh+1][n] }
//   C/D: vgpr v: row = v + 8*kh, col = lane&15
// ---------------------------------------------------------------------------
template <int K, int N>
__global__ void gcn_gemm_wmma(const float* __restrict__ A,
                              const float* __restrict__ W,
                              float* __restrict__ H) {
    constexpr int MT       = 4;
    constexpr int ROWS     = 16 * MT;          // 64
    constexpr int WAVES    = N / 16;
    constexpr int NTHREADS = WAVES * 32;
    constexpr int WSTRIDE  = N + WP;
    constexpr int ASTRIDE  = K + AP;

    extern __shared__ float smem[];
    float* Ws = smem;                          // [K][N+WP]
    float* As = smem + (size_t)K * WSTRIDE;    // [ROWS][K+AP]

    const int tid   = threadIdx.x;
    const int mbase = blockIdx.x * ROWS;

    // ---- stage W (K x N) into padded LDS (b128 global loads + ds stores) ----
    constexpr int wvecs = (K * N) / 4;
    for (int v = tid; v < wvecs; v += NTHREADS) {
        const int off = v * 4;
        const int r = off / N;
        const int c = off % N;
        *(v4f*)&Ws[r * WSTRIDE + c] = *(const v4f*)&W[(size_t)r * N + c];
    }

    // ---- stage A strip (ROWS x K) via CDNA5 async global->LDS b128 ----
    constexpr int avecs = (ROWS * K) / 4;
    for (int v = tid; v < avecs; v += NTHREADS) {
        const int off = v * 4;
        const int r = off / K;
        const int c = off % K;
        if (mbase + r < N_NODES) {
            const unsigned lds_off = (unsigned)(uintptr_t)&As[r * ASTRIDE + c];
            async_copy_b128(lds_off, &A[(size_t)(mbase + r) * K + c]);
        }
    }
    wait_asynccnt0();
    __syncthreads();

    const int lane = tid & 31;
    const int wave = tid >> 5;
    const int n0   = wave * 16;
    const int m    = lane & 15;   // A-row / B-col / D-col index within tile
    const int kh   = lane >> 4;

    v8f acc[MT] = {};
#pragma unroll 2
    for (int k = 0; k < K; k += 4) {
        v2f b;
        b.x = Ws[(k + 2 * kh) * WSTRIDE + n0 + m];
        b.y = Ws[(k + 2 * kh + 1) * WSTRIDE + n0 + m];
#pragma unroll
        for (int t = 0; t < MT; ++t) {
            v2f a = *(const v2f*)&As[(t * 16 + m) * ASTRIDE + k + 2 * kh];
            acc[t] = __builtin_amdgcn_wmma_f32_16x16x4_f32(
                /*neg_a=*/false, a, /*neg_b=*/false, b,
                /*c_mod=*/(short)0, acc[t], /*reuse_a=*/false, /*reuse_b=*/false);
        }
    }

#pragma unroll
    for (int t = 0; t < MT; ++t) {
        const int rbase = mbase + t * 16;
        if (rbase < N_NODES) {             // uniform per wave: EXEC stays all-1
#pragma unroll
            for (int v = 0; v < 8; ++v) {
                H[(size_t)(rbase + v + 8 * kh) * N + n0 + m] = acc[t][v];
            }
        }
    }
}

// ---------------------------------------------------------------------------
// Edge scatter:  Agg[col[e], :] += norm[e] * H[row[e], :]
// 8 edges per block; each thread: b128 gather + 4x global_atomic_add_f32.
// ---------------------------------------------------------------------------
template <int F>
__global__ void gcn_scatter(const float* __restrict__ H,
                            const int* __restrict__ row,
                            const int* __restrict__ col,
                            const float* __restrict__ norm,
                            float* __restrict__ Agg,
                            int n_edges) {
    const int e = blockIdx.x * 8 + threadIdx.y;
    if (e >= n_edges) return;
    const int   r = row[e];
    const int   c = col[e];
    const float w = norm[e];

    const int f = threadIdx.x * 4;                       // threadIdx.x in [0, F/4)
    v4f h = *(const v4f*)&H[(size_t)r * F + f];
    float* dst = &Agg[(size_t)c * F + f];
    atomicAdd(dst + 0, w * h.x);
    atomicAdd(dst + 1, w * h.y);
    atomicAdd(dst + 2, w * h.z);
    atomicAdd(dst + 3, w * h.w);
}

// ---------------------------------------------------------------------------
// Elementwise bias (+ optional ReLU).  F must be a power of two.
// ---------------------------------------------------------------------------
template <int F, bool RELU>
__global__ void gcn_bias_act(const float* __restrict__ In,
                             const float* __restrict__ bias,
                             float* __restrict__ Out,
                             int total) {
    const int i = blockIdx.x * blockDim.x + threadIdx.x;
    if (i >= total) return;
    float v = In[i] + bias[i & (F - 1)];
    if (RELU) v = fmaxf(v, 0.0f);
    Out[i] = v;
}

extern "C" void kernel_launch(void* const* d_in, const int* in_sizes, int n_in,
                              void* d_out, int out_size, void* d_ws, size_t ws_size,
                              hipStream_t stream) {
    const float* x    = (const float*)d_in[0];
    const int*   ei   = (const int*)d_in[1];    // [2, E] int
    const float* norm = (const float*)d_in[2];
    const float* W1   = (const float*)d_in[3];
    const float* b1   = (const float*)d_in[4];
    const float* W2   = (const float*)d_in[5];
    const float* b2   = (const float*)d_in[6];
    float*       out  = (float*)d_out;

    const int n_edges = in_sizes[1] / 2;        // 550000
    const int* row = ei;
    const int* col = ei + n_edges;

    // Workspace: two 50000x128 fp32 buffers (25.6 MB each), reused across layers.
    float* bufA = (float*)d_ws;                         // H1, then H1r (post-ReLU)
    float* bufB = bufA + (size_t)N_NODES * HID;         // Agg1, then H2

    const int strips = (N_NODES + 63) / 64;             // 782

    // ---- Layer 1:  H1 = x @ W1 ----
    {
        const size_t shmem = ((size_t)FEAT * (HID + WP) + (size_t)64 * (FEAT + AP)) * sizeof(float);
        gcn_gemm_wmma<FEAT, HID><<<strips, (HID / 16) * 32, shmem, stream>>>(x, W1, bufA);
    }

    hipMemsetAsync(bufB, 0, (size_t)N_NODES * HID * sizeof(float), stream);
    {
        dim3 blk(HID / 4, 8);
        int  grd = (n_edges + 7) / 8;
        gcn_scatter<HID><<<grd, blk, 0, stream>>>(bufA, row, col, norm, bufB, n_edges);
    }
    {
        const int total = N_NODES * HID;
        gcn_bias_act<HID, true><<<(total + 255) / 256, 256, 0, stream>>>(bufB, b1, bufA, total);
    }

    // ---- Layer 2:  H2 = relu(Agg1 + b1) @ W2 ----
    {
        const size_t shmem = ((size_t)HID * (EMB + WP) + (size_t)64 * (HID + AP)) * sizeof(float);
        gcn_gemm_wmma<HID, EMB><<<strips, (EMB / 16) * 32, shmem, stream>>>(bufA, W2, bufB);
    }

    hipMemsetAsync(out, 0, (size_t)N_NODES * EMB * sizeof(float), stream);
    {
        dim3 blk(EMB / 4, 8);
        int  grd = (n_edges + 7) / 8;
        gcn_scatter<EMB><<<grd, blk, 0, stream>>>(bufB, row, col, norm, out, n_edges);
    }
    {
        const int total = N_NODES * EMB;
        gcn_bias_act<EMB, false><<<(total + 255) / 256, 256, 0, stream>>>(out, b2, out, total);
    }
}